// QuantizedAttentionMap_14370960573293
// MI455X (gfx1250) — compile-verified
//
#include <hip/hip_runtime.h>
#include <math.h>

// ---------------------------------------------------------------------------
// QuantizedAttentionMap for MI455X (gfx1250).
// Memory-bound (≈1 GB traffic → ~45us @ 23.3 TB/s). No matmul structure, so
// the CDNA5 path used here is the async global->LDS DMA (ASYNCcnt) for the
// block-max pass, plus non-temporal stores to keep L2 for the re-read pass.
// ---------------------------------------------------------------------------

#define N_TEXT  226
#define N_IMG   5400
#define N_TOKEN 5626
#define HS      (N_TOKEN * N_TOKEN)   // 31,651,876 floats per head

typedef __attribute__((ext_vector_type(2))) float f2;
typedef __attribute__((ext_vector_type(4))) float f4;

#if __has_builtin(__builtin_amdgcn_global_load_async_to_lds_b64)
  #define HAVE_ASYNC_LDS 1
  // Builtin takes typed pointers: AS(1)/AS(3) pointers to an 8-byte int vector.
  typedef __attribute__((__vector_size__(8))) int v2i;
  typedef __attribute__((address_space(1))) v2i gv2i;   // global
  typedef __attribute__((address_space(3))) v2i lv2i;   // LDS
#else
  #define HAVE_ASYNC_LDS 0
#endif

__device__ __forceinline__ void wait_async0() {
#if __has_builtin(__builtin_amdgcn_s_wait_asynccnt)
  __builtin_amdgcn_s_wait_asynccnt(0);
#else
  asm volatile("s_wait_asynccnt 0" ::: "memory");
#endif
}

// ---------------------------------------------------------------------------
// Per-block dynamic fake-quant. One 256-thread workgroup (8 wave32s) per
// (head, block-row, block-col). Phase 1 streams the block through a 10-row
// LDS tile via async DMA and computes the block max; phase 2 re-reads the
// block from global (L2-resident, block <= 160KB << 192MB L2), quantizes and
// writes with NT stores.
// Head h -> nb in {27,36,45,54}, bw = 5400/nb in {200,150,120,100}.
// Block prefix sums: 729, 2025, 4050, 6966.
// ---------------------------------------------------------------------------
__launch_bounds__(256)
__global__ void qmap_block_quant(const float* __restrict__ x,
                                 float* __restrict__ out) {
  __shared__ __align__(16) float tile[2000];   // 10 rows x bw<=200 floats
  __shared__ float sred[8];
  __shared__ float sdelta;

  int b = (int)blockIdx.x;
  int h, nb, bw;
  if      (b <  729) { h = 0; nb = 27; bw = 200;            }
  else if (b < 2025) { h = 1; nb = 36; bw = 150; b -=  729; }
  else if (b < 4050) { h = 2; nb = 45; bw = 120; b -= 2025; }
  else               { h = 3; nb = 54; bw = 100; b -= 4050; }
  const int bi = b / nb;
  const int bj = b - bi * nb;

  // All offsets below are even (HS, N_TOKEN, N_TEXT, bw all even) -> f2/b64
  // accesses are 8-byte aligned.
  const unsigned base = (unsigned)h * HS
                      + (unsigned)(N_TEXT + bi * bw) * N_TOKEN
                      + (unsigned)(N_TEXT + bj * bw);
  const float* bp = x + base;
  float*       op = out + base;

  const int tid = (int)threadIdx.x;
  const int T = 10;                 // gcd of {100,120,150,200}
  const int t2 = (T * bw) >> 1;     // float2 elements per tile
  const int ntiles = bw / T;

  float m = -3.402823466e38f;
  for (int t = 0; t < ntiles; ++t) {
    const float* rp = bp + t * T * N_TOKEN;
#if HAVE_ASYNC_LDS
    for (int k = tid; k < t2; k += 256) {
      const int e  = k << 1;
      const int rr = e / bw;
      const int c  = e - rr * bw;               // even; f2 stays in-row
      float* g = const_cast<float*>(rp + rr * N_TOKEN + c);
      __builtin_amdgcn_global_load_async_to_lds_b64(
          (gv2i*)g, (lv2i*)&tile[e], /*offset=*/0, /*cpol=*/0);
    }
    wait_async0();                  // each wave drains its own ASYNCcnt
#else
    for (int k = tid; k < t2; k += 256) {
      const int e  = k << 1;
      const int rr = e / bw;
      const int c  = e - rr * bw;
      *(f2*)&tile[e] = *(const f2*)(rp + rr * N_TOKEN + c);
    }
#endif
    __syncthreads();                // all waves' DMA complete -> LDS visible
    const int nf = T * bw;
    for (int j = tid; j < nf; j += 256) m = fmaxf(m, tile[j]);
    __syncthreads();                // done reading before next tile overwrite
  }

  // wave32 reduction, then cross-wave via LDS
  for (int off = 16; off > 0; off >>= 1)
    m = fmaxf(m, __shfl_xor(m, off, 32));
  if ((tid & 31) == 0) sred[tid >> 5] = m;
  __syncthreads();
  if (tid == 0) {
    float d = sred[0];
    #pragma unroll
    for (int i = 1; i < 8; ++i) d = fmaxf(d, sred[i]);
    sdelta = d;
  }
  __syncthreads();

  const float delta = sdelta;
  const float scale = delta * (1.0f / 15.0f);
  const float safe  = (scale == 0.0f) ? 1.0f : scale;
  const float inv   = 1.0f / safe;

  // Phase 2: re-read (L2 hit), quantize (RTE rint == jnp.round), NT store.
  const int n2 = (bw * bw) >> 1;
  for (int k = tid; k < n2; k += 256) {
    const int e  = k << 1;
    const int rr = e / bw;
    const int c  = e - rr * bw;
    const unsigned off = (unsigned)rr * N_TOKEN + (unsigned)c;
    f2 v = *(const f2*)(bp + off);
    f2 q;
    q.x = fminf(fmaxf(rintf(v.x * inv), 0.0f), 15.0f) * scale;
    q.y = fminf(fmaxf(rintf(v.y * inv), 0.0f), 15.0f) * scale;
    __builtin_nontemporal_store(q, (f2*)(op + off));
  }
}

// ---------------------------------------------------------------------------
// Pass-through copies of the text region (touch-once -> NT load + NT store).
// Top strip per head: rows [0,226) full width = 226*5626 floats, contiguous,
// length and per-head base both divisible by 4 floats -> float4.
// ---------------------------------------------------------------------------
__global__ void qmap_copy_top(const float* __restrict__ x,
                              float* __restrict__ out) {
  const unsigned per = (N_TEXT * N_TOKEN) / 4;     // 317,869 float4s
  unsigned i = blockIdx.x * 256u + threadIdx.x;
  if (i >= 4u * per) return;
  const unsigned h = i / per;
  const unsigned r = i - h * per;
  const unsigned off = h * (HS / 4) + r;
  f4 v = __builtin_nontemporal_load((const f4*)x + off);
  __builtin_nontemporal_store(v, (f4*)out + off);
}

// Left strip per head: rows [226,5626), cols [0,226). 226 floats/row, row
// bases even -> float2 (113 per row).
__global__ void qmap_copy_left(const float* __restrict__ x,
                               float* __restrict__ out) {
  const unsigned per = N_IMG * (N_TEXT / 2);       // 5400*113 = 610,200 f2s
  unsigned i = blockIdx.x * 256u + threadIdx.x;
  if (i >= 4u * per) return;
  const unsigned h   = i / per;
  const unsigned rem = i - h * per;
  const unsigned r   = rem / 113u;
  const unsigned c   = rem - r * 113u;
  const unsigned off2 = (h * HS + (N_TEXT + r) * N_TOKEN) / 2 + c;
  f2 v = __builtin_nontemporal_load((const f2*)x + off2);
  __builtin_nontemporal_store(v, (f2*)out + off2);
}

extern "C" void kernel_launch(void* const* d_in, const int* in_sizes, int n_in,
                              void* d_out, int out_size, void* d_ws, size_t ws_size,
                              hipStream_t stream) {
  (void)in_sizes; (void)n_in; (void)out_size; (void)d_ws; (void)ws_size;
  const float* x = (const float*)d_in[0];
  float* out = (float*)d_out;

  // Text-region copies (independent of quant kernel).
  {
    const unsigned n = 4u * ((N_TEXT * N_TOKEN) / 4);      // 1,271,476
    qmap_copy_top<<<(n + 255u) / 256u, 256, 0, stream>>>(x, out);
  }
  {
    const unsigned n = 4u * (N_IMG * (N_TEXT / 2));        // 2,440,800
    qmap_copy_left<<<(n + 255u) / 256u, 256, 0, stream>>>(x, out);
  }
  // 729 + 1296 + 2025 + 2916 = 6966 blocks, one per quant block.
  qmap_block_quant<<<6966, 256, 0, stream>>>(x, out);
}